// FieldLoss_2345052144255
// MI455X (gfx1250) — compile-verified
//
#include <hip/hip_runtime.h>
#include <hip/hip_bf16.h>
#include <math.h>

// Problem constants (match reference setup_inputs)
#define BB 16
#define CC 16
#define HH 512
#define WW 512
#define PP (HH * WW)            // 262144 pixels per image
#define FMAX 1024
#define SLICES 16               // pixel slices per image -> 256 workgroups
#define TPB 256
#define PIX_PER_BLOCK (PP / SLICES)     // 16384
#define ITERS (PIX_PER_BLOCK / TPB)     // 64

// Workspace layout, in 32-bit words
#define WS_SUMS 0                                   // f32 [B*F*C]
#define WS_CNT  (BB * FMAX * CC)                    // u32 [B*F]
#define WS_HIST (WS_CNT + BB * FMAX)                // u32 [B*F*C]
#define WS_TOT  (WS_HIST + BB * FMAX * CC)          // f32 [2] : {sum_ce, sum_present}
#define WS_WORDS (WS_TOT + 2)

typedef __attribute__((ext_vector_type(2))) float v2f;
typedef __attribute__((ext_vector_type(8))) float v8f;

#if defined(__HIP_DEVICE_COMPILE__) && __has_builtin(__builtin_amdgcn_global_load_async_to_lds_b32)
#define ASYNC_COPY 1
#else
#define ASYNC_COPY 0
#endif

__device__ __forceinline__ void async_stage_b32(const int* gsrc, int* ldst) {
#if ASYNC_COPY
    __builtin_amdgcn_global_load_async_to_lds_b32(
        (__attribute__((address_space(1))) int*)(gsrc),
        (__attribute__((address_space(3))) int*)(ldst),
        0, 0);
#else
    (void)gsrc; (void)ldst;
#endif
}

template <int N>
__device__ __forceinline__ void wait_asynccnt_le() {
#if defined(__HIP_DEVICE_COMPILE__) && __has_builtin(__builtin_amdgcn_s_wait_asynccnt)
    __builtin_amdgcn_s_wait_asynccnt(N);
#elif defined(__gfx1250__)
    if constexpr (N == 0) {
        asm volatile("s_wait_asynccnt 0" ::: "memory");
    } else {
        asm volatile("s_wait_asynccnt 2" ::: "memory");
    }
#endif
}

// ---------------------------------------------------------------------------
// Kernel 0: zero the workspace (must run every launch; harness poisons once)
// ---------------------------------------------------------------------------
__global__ void fieldce_zero(unsigned int* __restrict__ ws, int n) {
    int i = blockIdx.x * blockDim.x + threadIdx.x;
    if (i < n) ws[i] = 0u;
}

// ---------------------------------------------------------------------------
// Kernel 1: per-slice scatter accumulation into LDS, sparse flush to L2 table
// ---------------------------------------------------------------------------
__global__ void fieldce_accum(const float* __restrict__ logits,
                              const int* __restrict__ masks,
                              const int* __restrict__ field_ids,
                              float* __restrict__ ws) {
    extern __shared__ char smem[];
    float*        s_sums = (float*)smem;                                    // F*C f32
    unsigned int* s_cnt  = (unsigned int*)(smem + FMAX * CC * 4);           // F   u32
    unsigned int* s_hist = (unsigned int*)(smem + (FMAX * CC + FMAX) * 4);  // F*C u32
    int*          s_fid  = (int*)(smem + (2 * FMAX * CC + FMAX) * 4);       // 2*TPB
    int*          s_lab  = s_fid + 2 * TPB;                                 // 2*TPB

    const int tid   = threadIdx.x;
    const int b     = blockIdx.x / SLICES;
    const int slice = blockIdx.x % SLICES;
    const int p0    = slice * PIX_PER_BLOCK;

    for (int i = tid; i < FMAX * CC; i += TPB) { s_sums[i] = 0.0f; s_hist[i] = 0u; }
    for (int i = tid; i < FMAX; i += TPB) s_cnt[i] = 0u;
    __syncthreads();

    const int*   gfid = field_ids + b * PP + p0;
    const int*   glab = masks     + b * PP + p0;
    const float* glog = logits + (size_t)b * CC * PP + p0;

#if ASYNC_COPY
    // prime double buffer 0
    async_stage_b32(gfid + tid, s_fid + tid);
    async_stage_b32(glab + tid, s_lab + tid);
#endif

    for (int it = 0; it < ITERS; ++it) {
        const int p   = it * TPB + tid;
        int fid, lab;
#if ASYNC_COPY
        const int buf  = it & 1;
        const int nbuf = buf ^ 1;
        if (it + 1 < ITERS) {
            async_stage_b32(gfid + (it + 1) * TPB + tid, s_fid + nbuf * TPB + tid);
            async_stage_b32(glab + (it + 1) * TPB + tid, s_lab + nbuf * TPB + tid);
            wait_asynccnt_le<2>();   // current tile's 2 loads retired (in-order)
        } else {
            wait_asynccnt_le<0>();
        }
        fid = s_fid[buf * TPB + tid];
        lab = s_lab[buf * TPB + tid];
#else
        fid = gfid[p];
        lab = glab[p];
#endif
        atomicAdd(&s_cnt[fid], 1u);
        atomicAdd(&s_hist[fid * CC + lab], 1u);
        const float* lp   = glog + p;
        float*       srow = s_sums + fid * CC;
#pragma unroll
        for (int c = 0; c < CC; ++c) {
            unsafeAtomicAdd(&srow[c], lp[(size_t)c * PP]);   // ds_add_f32
        }
    }
    __syncthreads();

    // Sparse flush of LDS partials to L2-resident global table
    float*        g_sums = ws + WS_SUMS + b * FMAX * CC;
    unsigned int* g_cnt  = (unsigned int*)ws + WS_CNT  + b * FMAX;
    unsigned int* g_hist = (unsigned int*)ws + WS_HIST + b * FMAX * CC;
    for (int i = tid; i < FMAX * CC; i += TPB) {
        float v = s_sums[i];
        if (v != 0.0f) unsafeAtomicAdd(&g_sums[i], v);
        unsigned int hv = s_hist[i];
        if (hv) atomicAdd(&g_hist[i], hv);
    }
    for (int i = tid; i < FMAX; i += TPB) {
        unsigned int cv = s_cnt[i];
        if (cv) atomicAdd(&g_cnt[i], cv);
    }
}

// ---------------------------------------------------------------------------
// Kernel 2: per-field CE + presence; WMMA-based block reduction
// ---------------------------------------------------------------------------
__global__ void fieldce_finalize(const float* __restrict__ ws, float* __restrict__ totals) {
    const unsigned int* ws_u = (const unsigned int*)ws;
    const int idx = blockIdx.x * TPB + threadIdx.x;       // 0 .. B*F-1
    const int f   = idx % FMAX;

    const float*        sums = ws + WS_SUMS + idx * CC;
    const unsigned int  cnt  = ws_u[WS_CNT + idx];
    const unsigned int* hist = ws_u + WS_HIST + idx * CC;

    const float inv = 1.0f / fmaxf((float)cnt, 1.0f);

    // stable logsumexp over mean logits
    float mx = -INFINITY;
#pragma unroll
    for (int c = 0; c < CC; ++c) mx = fmaxf(mx, sums[c] * inv);
    float se = 0.0f;
#pragma unroll
    for (int c = 0; c < CC; ++c) se += expf(sums[c] * inv - mx);
    const float logz = mx + logf(se);

    // histogram mode: first max == smallest index among maxima (argmax semantics)
    unsigned int bestAll = hist[0]; int modeAll = 0;
    unsigned int bestVal = hist[1]; int modeVal = 1;
    unsigned int validCnt = 0;
#pragma unroll
    for (int c = 1; c < CC; ++c) {
        unsigned int h = hist[c];
        validCnt += h;
        if (h > bestVal) { bestVal = h; modeVal = c; }
        if (h > bestAll) { bestAll = h; modeAll = c; }
    }
    const int   label  = (validCnt > 0u) ? modeVal : modeAll;
    const float picked = sums[label] * inv;
    const float ce     = logz - picked;
    const float w      = (cnt > 0u && f != 0) ? 1.0f : 0.0f;

    __shared__ float rbuf[2 * TPB];
    rbuf[threadIdx.x]       = ce * w;
    rbuf[TPB + threadIdx.x] = w;
    __syncthreads();

    // Wave 0 reduces both 256-value arrays with chained v_wmma_f32_16x16x4_f32:
    // A = all-ones 16x4, B = 4x16 tiles of data. With A==1, sum_n D[0][n] is the
    // exact f32 grand total regardless of the B operand's lane layout, since each
    // value occupies exactly one B slot across the 4 chained calls.
    if (threadIdx.x < 32) {
        const int l = threadIdx.x;
        v2f ones; ones.x = 1.0f; ones.y = 1.0f;
        float tot[2];
#pragma unroll
        for (int r = 0; r < 2; ++r) {
            const float* src = rbuf + r * TPB;
            v8f acc = {};
#pragma unroll
            for (int j = 0; j < 4; ++j) {
                v2f bm;
                bm.x = src[j * 64 + l];
                bm.y = src[j * 64 + 32 + l];
                acc = __builtin_amdgcn_wmma_f32_16x16x4_f32(
                    false, ones, false, bm, (short)0, acc, false, false);
            }
            float v = acc[0];             // column sum for N = lane%16 (M-replicated)
            v += __shfl_xor(v, 1, 16);
            v += __shfl_xor(v, 2, 16);
            v += __shfl_xor(v, 4, 16);
            v += __shfl_xor(v, 8, 16);
            tot[r] = v;                    // lane 0: sum of all 16 column sums
        }
        if (l == 0) {
            unsafeAtomicAdd(&totals[0], tot[0]);
            unsafeAtomicAdd(&totals[1], tot[1]);
        }
    }
}

// ---------------------------------------------------------------------------
// Kernel 3: final scalar
// ---------------------------------------------------------------------------
__global__ void fieldce_writeout(const float* __restrict__ totals, float* __restrict__ out) {
    const float t = totals[0];
    const float n = totals[1];
    out[0] = (n > 0.0f) ? (t / fmaxf(n, 1.0f)) : 0.0f;
}

// ---------------------------------------------------------------------------
extern "C" void kernel_launch(void* const* d_in, const int* in_sizes, int n_in,
                              void* d_out, int out_size, void* d_ws, size_t ws_size,
                              hipStream_t stream) {
    (void)in_sizes; (void)n_in; (void)out_size; (void)ws_size;

    const float* logits    = (const float*)d_in[0];
    const int*   masks     = (const int*)d_in[1];
    const int*   field_ids = (const int*)d_in[2];
    float*       ws        = (float*)d_ws;
    float*       totals    = ws + WS_TOT;

    const size_t lds_bytes = (size_t)(2 * FMAX * CC + FMAX) * 4 + (size_t)4 * TPB * 4; // 139264

    // Allow >64KB dynamic LDS (host-side attribute; not a stream op, capture-safe)
    (void)hipFuncSetAttribute((const void*)fieldce_accum,
                              hipFuncAttributeMaxDynamicSharedMemorySize, (int)lds_bytes);

    fieldce_zero<<<(WS_WORDS + TPB - 1) / TPB, TPB, 0, stream>>>((unsigned int*)d_ws, WS_WORDS);
    fieldce_accum<<<BB * SLICES, TPB, lds_bytes, stream>>>(logits, masks, field_ids, ws);
    fieldce_finalize<<<(BB * FMAX) / TPB, TPB, 0, stream>>>(ws, totals);
    fieldce_writeout<<<1, 1, 0, stream>>>(totals, (float*)d_out);
}